// SchNetLikeModel_4329327034535
// MI455X (gfx1250) — compile-verified
//
#include <hip/hip_runtime.h>
#include <hip/hip_bf16.h>

// ---------------------------------------------------------------------------
// SchNet-like GNN on MI455X (gfx1250, wave32).
// fp32 WMMA (V_WMMA_F32_16X16X4_F32) for the per-node MLP GEMMs; the edge
// scatter (memory/atomic bound, ~1.7 GB total traffic) is a float4 gather +
// global_atomic_add_f32. GraphNorm exploits the sorted batch array: register
// accumulation per feature lane with atomic flush only on graph transitions.
// ---------------------------------------------------------------------------

#define N_NODES  20000
#define N_EDGES  320000
#define N_FEAT   128
#define MLP_HID  64
#define N_GRAPHS 16
#define N_LAYERS 5
#define GN_EPS   1e-5f

typedef __attribute__((ext_vector_type(2))) float v2f;
typedef __attribute__((ext_vector_type(8))) float v8f;

#define HSTR 65  // padded LDS stride (odd -> conflict-free column reads)

// ---------------------------------------------------------------------------
// Per-node MLP: m = relu(h @ W1 + b1) @ W2 + b2 ; also acc = m (self loop).
// 1 wave (32 threads) per 16-row tile; 20000/16 = 1250 tiles exactly.
// ---------------------------------------------------------------------------
__global__ __launch_bounds__(32)
void mlp_kernel(const float* __restrict__ hin,
                const float* __restrict__ W1, const float* __restrict__ b1,
                const float* __restrict__ W2, const float* __restrict__ b2,
                float* __restrict__ mbuf, float* __restrict__ accbuf)
{
    __shared__ float hid[16 * HSTR];

    const int lane = threadIdx.x & 31;
    const int half = lane >> 4;        // 0: K lo pair, 1: K hi pair
    const int l16  = lane & 15;
    const int row0 = blockIdx.x * 16;

    // ---------------- GEMM1: [16x128] @ [128x64] -> 16x64 hidden ----------
    v8f c0 = {}, c1 = {}, c2 = {}, c3 = {};
    const float* arow = hin + (size_t)(row0 + l16) * N_FEAT;

    #pragma unroll 4
    for (int k0 = 0; k0 < N_FEAT; k0 += 4) {
        const int ka = k0 + half * 2;
        v2f a; a.x = arow[ka]; a.y = arow[ka + 1];

        const float* w = W1 + (size_t)ka * MLP_HID + l16;
        v2f b;
        b.x = w[0];              b.y = w[MLP_HID];
        c0 = __builtin_amdgcn_wmma_f32_16x16x4_f32(false, a, false, b, (short)0, c0, false, false);
        b.x = w[16];             b.y = w[MLP_HID + 16];
        c1 = __builtin_amdgcn_wmma_f32_16x16x4_f32(false, a, false, b, (short)0, c1, false, false);
        b.x = w[32];             b.y = w[MLP_HID + 32];
        c2 = __builtin_amdgcn_wmma_f32_16x16x4_f32(false, a, false, b, (short)0, c2, false, false);
        b.x = w[48];             b.y = w[MLP_HID + 48];
        c3 = __builtin_amdgcn_wmma_f32_16x16x4_f32(false, a, false, b, (short)0, c3, false, false);
    }

    // bias + relu, stage hidden tile to LDS (C layout -> row major)
    {
        v8f cc[4] = {c0, c1, c2, c3};
        #pragma unroll
        for (int n = 0; n < 4; ++n) {
            const int   col  = n * 16 + l16;
            const float bias = b1[col];
            #pragma unroll
            for (int r = 0; r < 8; ++r) {
                const int mrow = r + half * 8;
                hid[mrow * HSTR + col] = fmaxf(cc[n][r] + bias, 0.0f);
            }
        }
    }
    __syncthreads();   // single wave: compiles to s_wait_dscnt ordering

    // ---------------- GEMM2: [16x64] @ [64x128] -> 16x128 -----------------
    v8f d[8] = {};
    #pragma unroll 4
    for (int k0 = 0; k0 < MLP_HID; k0 += 4) {
        const int ka = k0 + half * 2;
        v2f a;
        a.x = hid[l16 * HSTR + ka];
        a.y = hid[l16 * HSTR + ka + 1];

        const float* w = W2 + (size_t)ka * N_FEAT + l16;
        #pragma unroll
        for (int j = 0; j < 8; ++j) {
            v2f b;
            b.x = w[j * 16];
            b.y = w[N_FEAT + j * 16];
            d[j] = __builtin_amdgcn_wmma_f32_16x16x4_f32(false, a, false, b, (short)0, d[j], false, false);
        }
    }

    // bias, write m and acc (= m, the self-loop contribution)
    #pragma unroll
    for (int j = 0; j < 8; ++j) {
        const int   col  = j * 16 + l16;
        const float bias = b2[col];
        #pragma unroll
        for (int r = 0; r < 8; ++r) {
            const int    grow = row0 + r + half * 8;
            const float  v    = d[j][r] + bias;
            const size_t idx  = (size_t)grow * N_FEAT + col;
            mbuf[idx]   = v;
            accbuf[idx] = v;
        }
    }
}

// ---------------------------------------------------------------------------
// Edge scatter: acc[dst] += m[src]. One thread per (edge, 4-feature chunk).
// ---------------------------------------------------------------------------
__global__ __launch_bounds__(256)
void scatter_kernel(const float* __restrict__ mbuf,
                    const int* __restrict__ src, const int* __restrict__ dst,
                    float* __restrict__ accbuf)
{
    const size_t gid = (size_t)blockIdx.x * blockDim.x + threadIdx.x;
    const int e  = (int)(gid >> 5);
    const int f4 = ((int)gid & 31) * 4;
    if (e >= N_EDGES) return;

    const int s = src[e];
    const int t = dst[e];
    const float4 v = *(const float4*)(mbuf + (size_t)s * N_FEAT + f4);
    float* p = accbuf + (size_t)t * N_FEAT + f4;
    atomicAdd(p + 0, v.x);
    atomicAdd(p + 1, v.y);
    atomicAdd(p + 2, v.z);
    atomicAdd(p + 3, v.w);
}

// ---------------------------------------------------------------------------
// Per-graph segment sum (batch sorted). 128 threads = one feature each,
// 64 nodes per block; flush on graph transition only.
// ---------------------------------------------------------------------------
__global__ __launch_bounds__(128)
void seg_sum_kernel(const float* __restrict__ h, const int* __restrict__ batch,
                    float* __restrict__ sums)
{
    const int f  = threadIdx.x;
    const int n0 = blockIdx.x * 64;
    const int n1 = min(n0 + 64, N_NODES);
    float a   = 0.0f;
    int   cur = batch[n0];
    for (int n = n0; n < n1; ++n) {
        const int g = batch[n];
        if (g != cur) {
            atomicAdd(&sums[cur * N_FEAT + f], a);
            a = 0.0f; cur = g;
        }
        a += h[(size_t)n * N_FEAT + f];
    }
    atomicAdd(&sums[cur * N_FEAT + f], a);
}

// ---------------------------------------------------------------------------
// GraphNorm pass 2: out = h - gn_scale * mean[batch] (in place) and
// accumulate per-graph sum of out^2.
// ---------------------------------------------------------------------------
__global__ __launch_bounds__(128)
void center_kernel(float* __restrict__ acc, const int* __restrict__ batch,
                   const float* __restrict__ sums, const float* __restrict__ cnt,
                   const float* __restrict__ gscale, float* __restrict__ sqsums)
{
    const int f  = threadIdx.x;
    const int n0 = blockIdx.x * 64;
    const int n1 = min(n0 + 64, N_NODES);
    const float al = gscale[f];

    int   cur  = batch[n0];
    float mean = sums[cur * N_FEAT + f] / fmaxf(cnt[cur], 1.0f);
    float sq   = 0.0f;
    for (int n = n0; n < n1; ++n) {
        const int g = batch[n];
        if (g != cur) {
            atomicAdd(&sqsums[cur * N_FEAT + f], sq);
            sq = 0.0f; cur = g;
            mean = sums[g * N_FEAT + f] / fmaxf(cnt[g], 1.0f);
        }
        const size_t idx = (size_t)n * N_FEAT + f;
        const float  v   = acc[idx] - al * mean;
        acc[idx] = v;
        sq += v * v;
    }
    atomicAdd(&sqsums[cur * N_FEAT + f], sq);
}

// ---------------------------------------------------------------------------
// GraphNorm pass 3: normalize, affine, relu, residual -> h.
// ---------------------------------------------------------------------------
__global__ __launch_bounds__(256)
void norm_kernel(const float* __restrict__ centered, const int* __restrict__ batch,
                 const float* __restrict__ sqsums, const float* __restrict__ cnt,
                 const float* __restrict__ w, const float* __restrict__ b,
                 float* __restrict__ h, int layer)
{
    const size_t gid = (size_t)blockIdx.x * blockDim.x + threadIdx.x;
    if (gid >= (size_t)N_NODES * N_FEAT) return;
    const int n = (int)(gid >> 7);
    const int f = (int)gid & (N_FEAT - 1);
    const int g = batch[n];
    const float var = sqsums[g * N_FEAT + f] / fmaxf(cnt[g], 1.0f);
    float v = centered[gid] * rsqrtf(var + GN_EPS);
    v = fmaxf(w[f] * v + b[f], 0.0f);
    h[gid] = (layer == 0) ? v : v + h[gid];
}

// ---------------------------------------------------------------------------
// Small helpers: zero fill and per-graph node counts.
// ---------------------------------------------------------------------------
__global__ void zero_kernel(float* __restrict__ p, int n)
{
    const int i = blockIdx.x * blockDim.x + threadIdx.x;
    if (i < n) p[i] = 0.0f;
}

__global__ void count_kernel(const int* __restrict__ batch, float* __restrict__ cnt)
{
    const int n = blockIdx.x * blockDim.x + threadIdx.x;
    if (n < N_NODES) atomicAdd(&cnt[batch[n]], 1.0f);
}

// ---------------------------------------------------------------------------
// Readout: out[g] = dot(sums[g]/cnt[g], lin_w) + lin_b. One block per graph.
// ---------------------------------------------------------------------------
__global__ __launch_bounds__(128)
void readout_kernel(const float* __restrict__ sums, const float* __restrict__ cnt,
                    const float* __restrict__ lin_w, const float* __restrict__ lin_b,
                    float* __restrict__ out)
{
    __shared__ float red[N_FEAT];
    const int g = blockIdx.x;
    const int f = threadIdx.x;
    red[f] = (sums[g * N_FEAT + f] / fmaxf(cnt[g], 1.0f)) * lin_w[f];
    __syncthreads();
    #pragma unroll
    for (int s = 64; s > 0; s >>= 1) {
        if (f < s) red[f] += red[f + s];
        __syncthreads();
    }
    if (f == 0) out[g] = red[0] + lin_b[0];
}

// ---------------------------------------------------------------------------
extern "C" void kernel_launch(void* const* d_in, const int* in_sizes, int n_in,
                              void* d_out, int out_size, void* d_ws, size_t ws_size,
                              hipStream_t stream)
{
    const float* x         = (const float*)d_in[0];
    const int*   edges     = (const int*)d_in[1];        // [2, N_EDGES]
    const int*   batch     = (const int*)d_in[2];        // [N_NODES], sorted
    const float* W1        = (const float*)d_in[3];      // [L,128,64]
    const float* b1        = (const float*)d_in[4];      // [L,64]
    const float* W2        = (const float*)d_in[5];      // [L,64,128]
    const float* b2        = (const float*)d_in[6];      // [L,128]
    const float* gn_weight = (const float*)d_in[7];      // [L,128]
    const float* gn_bias   = (const float*)d_in[8];      // [L,128]
    const float* gn_scale  = (const float*)d_in[9];      // [L,128]
    const float* lin_w     = (const float*)d_in[10];     // [128]
    const float* lin_b     = (const float*)d_in[11];     // [1]
    float*       out       = (float*)d_out;

    const int* src = edges;
    const int* dst = edges + N_EDGES;

    // workspace layout (floats)
    float* h    = (float*)d_ws;                          // [N,128]
    float* m    = h    + (size_t)N_NODES * N_FEAT;       // [N,128]
    float* acc  = m    + (size_t)N_NODES * N_FEAT;       // [N,128]
    float* sums = acc  + (size_t)N_NODES * N_FEAT;       // [16,128]
    float* sq   = sums + N_GRAPHS * N_FEAT;              // [16,128] (contiguous!)
    float* cnt  = sq   + N_GRAPHS * N_FEAT;              // [16]

    const int segBlocks  = (N_NODES + 63) / 64;                       // 313
    const int scatBlocks = (int)(((size_t)N_EDGES * 32 + 255) / 256); // 40000
    const int elemBlocks = (int)(((size_t)N_NODES * N_FEAT + 255) / 256);

    // per-graph node counts (fresh each call: deterministic)
    zero_kernel<<<1, 32, 0, stream>>>(cnt, N_GRAPHS);
    count_kernel<<<(N_NODES + 255) / 256, 256, 0, stream>>>(batch, cnt);

    for (int i = 0; i < N_LAYERS; ++i) {
        const float* hin = (i == 0) ? x : h;

        mlp_kernel<<<N_NODES / 16, 32, 0, stream>>>(
            hin,
            W1 + (size_t)i * N_FEAT * MLP_HID, b1 + (size_t)i * MLP_HID,
            W2 + (size_t)i * MLP_HID * N_FEAT, b2 + (size_t)i * N_FEAT,
            m, acc);

        scatter_kernel<<<scatBlocks, 256, 0, stream>>>(m, src, dst, acc);

        // zero sums+sq (contiguous 2*16*128 floats)
        zero_kernel<<<(2 * N_GRAPHS * N_FEAT + 255) / 256, 256, 0, stream>>>(
            sums, 2 * N_GRAPHS * N_FEAT);

        seg_sum_kernel<<<segBlocks, 128, 0, stream>>>(acc, batch, sums);

        center_kernel<<<segBlocks, 128, 0, stream>>>(
            acc, batch, sums, cnt, gn_scale + (size_t)i * N_FEAT, sq);

        norm_kernel<<<elemBlocks, 256, 0, stream>>>(
            acc, batch, sq, cnt,
            gn_weight + (size_t)i * N_FEAT, gn_bias + (size_t)i * N_FEAT,
            h, i);
    }

    // mean pool + linear readout
    zero_kernel<<<(N_GRAPHS * N_FEAT + 255) / 256, 256, 0, stream>>>(
        sums, N_GRAPHS * N_FEAT);
    seg_sum_kernel<<<segBlocks, 128, 0, stream>>>(h, batch, sums);
    readout_kernel<<<N_GRAPHS, 128, 0, stream>>>(sums, cnt, lin_w, lin_b, out);
}